// LIF_67714454389107
// MI455X (gfx1250) — compile-verified
//
#include <hip/hip_runtime.h>

// LIF recurrence: x [B,T,N] f32 -> spikes [B,T,N] f32.
// Purely HBM-bound (AI ~0.65 flop/byte; ideal ~11 us at 23.3 TB/s).
// Strategy: CDNA5 async global->LDS pipeline (ASYNCcnt) for deep MLP at
// near-zero VGPR cost, register-resident membrane state, NT cache policy
// for the one-touch streaming data, wave32-sized slices.

constexpr int B_DIM   = 128;
constexpr int T_DIM   = 32;
constexpr int N_DIM   = 8192;

constexpr int NCHUNK  = 1024;         // floats per block per timestep (4 KB row slice)
constexpr int THREADS = NCHUNK / 4;   // 256 threads, one float4 each (8 wave32s)
constexpr int DEPTH   = 8;            // async pipeline depth (ring slots in LDS)

constexpr int T_STRIDE_BYTES = N_DIM * 4;        // 32768 B; t*stride < 2^23 -> fits offset imm
constexpr int SLOT_BYTES     = NCHUNK * 4;       // 4096 B per LDS ring slot

typedef float f32x4 __attribute__((ext_vector_type(4)));

// ---- CDNA5 async global->LDS copy (16B/lane), ASYNCcnt-tracked, NT hint ----
__device__ __forceinline__ void async_load_16B(unsigned lds_addr, const float* gbase,
                                               int byteoff) {
    // VDST = wave-relative LDS byte address, ADDR = 64-bit global base VGPR pair,
    // timestep selected via the signed 24-bit instruction offset.
    asm volatile("global_load_async_to_lds_b128 %0, %1, off offset:%2 th:TH_LOAD_NT"
                 :
                 : "v"(lds_addr), "v"(gbase), "n"(byteoff)
                 : "memory");
}

// Wait until outstanding async copies <= n (async loads complete in order).
__device__ __forceinline__ void wait_async_le(int n) {
    switch (n) {
    case 0: asm volatile("s_wait_asynccnt 0" ::: "memory"); break;
    case 1: asm volatile("s_wait_asynccnt 1" ::: "memory"); break;
    case 2: asm volatile("s_wait_asynccnt 2" ::: "memory"); break;
    case 3: asm volatile("s_wait_asynccnt 3" ::: "memory"); break;
    case 4: asm volatile("s_wait_asynccnt 4" ::: "memory"); break;
    case 5: asm volatile("s_wait_asynccnt 5" ::: "memory"); break;
    case 6: asm volatile("s_wait_asynccnt 6" ::: "memory"); break;
    default: asm volatile("s_wait_asynccnt 7" ::: "memory"); break;
    }
}

__device__ __forceinline__ void wait_ds0() {
    asm volatile("s_wait_dscnt 0" ::: "memory");
}

__global__ __launch_bounds__(THREADS) void lif_async_kernel(const float* __restrict__ x,
                                                            float* __restrict__ out) {
    __shared__ float smem[DEPTH * NCHUNK];

    const int tid = threadIdx.x;                       // 0..255
    const int b   = blockIdx.y;                        // 0..127
    const int n0  = blockIdx.x * NCHUNK + tid * 4;     // this thread's 4 floats

    // Low 32 bits of a generic shared-pointer == wave-relative LDS byte address
    const unsigned lds_lane =
        (unsigned)(unsigned long long)(const void*)&smem[tid * 4];

    const float* gbase = x   + (size_t)b * T_DIM * N_DIM + n0;  // t = 0 slice
    float*       obase = out + (size_t)b * T_DIM * N_DIM + n0;

    // Prologue: fill the pipeline (single base VGPR pair, immediate offsets).
#pragma unroll
    for (int t = 0; t < DEPTH; ++t) {
        async_load_16B(lds_lane + (unsigned)(t * SLOT_BYTES), gbase, t * T_STRIDE_BYTES);
    }

    f32x4 u = (f32x4)0.0f;

#pragma unroll
    for (int t = 0; t < T_DIM; ++t) {
        const int slot = t & (DEPTH - 1);

        // Ensure the copy for timestep t has landed in LDS (in-order completion).
        const int allow = (T_DIM - 1 - t) < (DEPTH - 1) ? (T_DIM - 1 - t) : (DEPTH - 1);
        wait_async_le(allow);

        const f32x4 xv = *(const f32x4*)&smem[slot * NCHUNK + tid * 4];

        // Guard the ring slot: the ds_load above must complete before the async
        // DMA for timestep t+DEPTH may overwrite it.
        wait_ds0();
        if (t + DEPTH < T_DIM) {
            async_load_16B(lds_lane + (unsigned)(slot * SLOT_BYTES), gbase,
                           (t + DEPTH) * T_STRIDE_BYTES);
        }

        // LIF update: u = u/2 + x; s = (u >= 1); u = s ? 0 : u.
        f32x4 s;
#pragma unroll
        for (int c = 0; c < 4; ++c) {
            const float uu  = u[c] * 0.5f + xv[c];
            const bool fire = (uu >= 1.0f);
            s[c] = fire ? 1.0f : 0.0f;
            u[c] = fire ? 0.0f : uu;
        }

        // One-touch streaming output: non-temporal 16B store.
        __builtin_nontemporal_store(s, (f32x4*)(obase + (size_t)t * N_DIM));
    }
}

extern "C" void kernel_launch(void* const* d_in, const int* in_sizes, int n_in,
                              void* d_out, int out_size, void* d_ws, size_t ws_size,
                              hipStream_t stream) {
    (void)in_sizes; (void)n_in; (void)out_size; (void)d_ws; (void)ws_size;
    const float* x = (const float*)d_in[0];
    float* out     = (float*)d_out;

    dim3 grid(N_DIM / NCHUNK, B_DIM);   // 8 x 128 = 1024 blocks
    lif_async_kernel<<<grid, THREADS, 0, stream>>>(x, out);
}